// QuantumAttentionMechanism_2396591751518
// MI455X (gfx1250) — compile-verified
//
#include <hip/hip_runtime.h>
#include <math.h>

// ---------------------------------------------------------------------------
// QuantumAttentionMechanism for MI455X (gfx1250, wave32).
// FP32 end-to-end using V_WMMA_F32_16X16X4_F32 for every GEMM (dominant cost:
// 3x batched 1024^3 fp32 GEMMs for the enhancement self-correlation, ~206 GF).
// Double-buffered LDS staging with async global->LDS copies
// (GLOBAL_LOAD_ASYNC_TO_LDS_B128 / ASYNCcnt); fallback register-split
// prefetch if the builtin is absent.
// ---------------------------------------------------------------------------

typedef __attribute__((ext_vector_type(2))) float v2f;
typedef __attribute__((ext_vector_type(8))) float v8f;
typedef __attribute__((ext_vector_type(4))) int   v4i;

#define TILE_M 128
#define TILE_N 64
#define KC     32
#define TS     36   // padded LDS row stride (floats) to dodge bank conflicts

#define AS1 __attribute__((address_space(1)))
#define AS3 __attribute__((address_space(3)))

#if defined(__has_builtin)
#if __has_builtin(__builtin_amdgcn_global_load_async_to_lds_b128)
#define CDNA5_ASYNC_LDS 1
#endif
#endif
#ifndef CDNA5_ASYNC_LDS
#define CDNA5_ASYNC_LDS 0
#endif

#if CDNA5_ASYNC_LDS
__device__ __forceinline__ void async_cp16(float* lds, const float* g) {
    // param0: AS1 (device/global) v4i*  [src],  param1: AS3 (LDS) v4i* [dst]
    __builtin_amdgcn_global_load_async_to_lds_b128(
        (AS1 v4i*)g, (AS3 v4i*)lds, 0, 0);
}
__device__ __forceinline__ void wait_async0() {
#if __has_builtin(__builtin_amdgcn_s_wait_asynccnt)
    __builtin_amdgcn_s_wait_asynccnt(0);
#else
    asm volatile("s_wait_asynccnt 0x0" ::: "memory");
#endif
}
#endif

// Generic batched GEMM:  C = alpha * A * op(B) (+ bias)
//   BT=true : op(B) = B^T with B stored [N][K] row-major  (NT gemm)
//   BT=false: op(B) = B   with B stored [K][N] row-major  (NN gemm)
// Batch index z = blockIdx.z decomposed as zb = z/heads, zh = z%heads with
// independent (outer, inner) element strides per operand.
template <bool BT>
__global__ __launch_bounds__(256) void gemm_wmma_f32(
    const float* __restrict__ A, int lda, long long saO, long long saI,
    const float* __restrict__ B, int ldb, long long sbO, long long sbI,
    float* __restrict__ C, int ldc, long long scO, long long scI,
    int M, int N, int K, int heads, float alpha, const float* __restrict__ bias)
{
    __shared__ __align__(16) float As[2][TILE_M * TS];
    __shared__ __align__(16) float Bs[2][TILE_N * TS];

    const int z  = blockIdx.z;
    const int zb = z / heads;
    const int zh = z - zb * heads;
    const float* Ab = A + (long long)zb * saO + (long long)zh * saI;
    const float* Bb = B + (long long)zb * sbO + (long long)zh * sbI;
    float*       Cb = C + (long long)zb * scO + (long long)zh * scI;

    const int row0 = blockIdx.y * TILE_M;
    const int col0 = blockIdx.x * TILE_N;

    const int tid  = threadIdx.x;
    const int lane = tid & 31;
    const int wave = tid >> 5;   // 0..7
    const int wm   = wave >> 1;  // 0..3 : 32-row band
    const int wn   = wave & 1;   // 0..1 : 32-col band
    const int half = lane >> 4;  // K-pair select per ISA f32 A/B layout
    const int lm   = lane & 15;

    const v8f zero8 = {0.f, 0.f, 0.f, 0.f, 0.f, 0.f, 0.f, 0.f};
    v8f acc[2][2];
    acc[0][0] = zero8; acc[0][1] = zero8; acc[1][0] = zero8; acc[1][1] = zero8;

    // Global->LDS tile load mappings (all sizes exact multiples; no guards).
    const int akc = (tid & 7) << 2;   // 0..28 : k within chunk (float4)
    const int am0 = tid >> 3;         // 0..31 : row within 32-row pass
    const int nnc = (tid & 15) << 2;  // NN: n within tile (float4)
    const int kk0 = tid >> 4;         // NN: k row within chunk

    int cur = 0;

#if CDNA5_ASYNC_LDS
    // ---------------- async-copy staging (ASYNCcnt path) ----------------
    {
        // prologue: stage chunk 0 into buffer 0
#pragma unroll
        for (int p = 0; p < 4; ++p) {
            const int m = am0 + p * 32;
            async_cp16(&As[0][m * TS + akc],
                       Ab + (long long)(row0 + m) * lda + akc);
        }
        if (BT) {
#pragma unroll
            for (int p = 0; p < 2; ++p) {
                const int n = am0 + p * 32;
                async_cp16(&Bs[0][n * TS + akc],
                           Bb + (long long)(col0 + n) * ldb + akc);
            }
        } else {
#pragma unroll
            for (int p = 0; p < 2; ++p) {
                const int k = kk0 + p * 16;
                const float4 v = *(const float4*)(Bb + (long long)k * ldb + col0 + nnc);
                Bs[0][(nnc + 0) * TS + k] = v.x;
                Bs[0][(nnc + 1) * TS + k] = v.y;
                Bs[0][(nnc + 2) * TS + k] = v.z;
                Bs[0][(nnc + 3) * TS + k] = v.w;
            }
        }
        wait_async0();
        __syncthreads();
    }

    for (int kt = 0; kt < K; kt += KC) {
        const bool more = (kt + KC) < K;
        if (more) {
            const int nxt = cur ^ 1;
            const int kn  = kt + KC;
#pragma unroll
            for (int p = 0; p < 4; ++p) {
                const int m = am0 + p * 32;
                async_cp16(&As[nxt][m * TS + akc],
                           Ab + (long long)(row0 + m) * lda + kn + akc);
            }
            if (BT) {
#pragma unroll
                for (int p = 0; p < 2; ++p) {
                    const int n = am0 + p * 32;
                    async_cp16(&Bs[nxt][n * TS + akc],
                               Bb + (long long)(col0 + n) * ldb + kn + akc);
                }
            } else {
#pragma unroll
                for (int p = 0; p < 2; ++p) {
                    const int k = kk0 + p * 16;
                    const float4 v =
                        *(const float4*)(Bb + (long long)(kn + k) * ldb + col0 + nnc);
                    Bs[nxt][(nnc + 0) * TS + k] = v.x;
                    Bs[nxt][(nnc + 1) * TS + k] = v.y;
                    Bs[nxt][(nnc + 2) * TS + k] = v.z;
                    Bs[nxt][(nnc + 3) * TS + k] = v.w;
                }
            }
        }

        // ---- compute on current buffer: 8 K-steps of 4, 2x2 subtiles ----
        {
            const float* __restrict__ Asb = As[cur];
            const float* __restrict__ Bsb = Bs[cur];
#pragma unroll
            for (int kk = 0; kk < KC; kk += 4) {
                const int ko = kk + 2 * half;  // even -> 8B aligned v2f loads
                const v2f a0 = *(const v2f*)&Asb[(wm * 32 +  0 + lm) * TS + ko];
                const v2f a1 = *(const v2f*)&Asb[(wm * 32 + 16 + lm) * TS + ko];
                const v2f b0 = *(const v2f*)&Bsb[(wn * 32 +  0 + lm) * TS + ko];
                const v2f b1 = *(const v2f*)&Bsb[(wn * 32 + 16 + lm) * TS + ko];
                acc[0][0] = __builtin_amdgcn_wmma_f32_16x16x4_f32(
                    false, a0, false, b0, (short)0, acc[0][0], false, false);
                acc[0][1] = __builtin_amdgcn_wmma_f32_16x16x4_f32(
                    false, a0, false, b1, (short)0, acc[0][1], false, false);
                acc[1][0] = __builtin_amdgcn_wmma_f32_16x16x4_f32(
                    false, a1, false, b0, (short)0, acc[1][0], false, false);
                acc[1][1] = __builtin_amdgcn_wmma_f32_16x16x4_f32(
                    false, a1, false, b1, (short)0, acc[1][1], false, false);
            }
        }

        if (more) {
            wait_async0();
            __syncthreads();
            cur ^= 1;
        }
    }
#else
    // ------- fallback: register-split prefetch + double LDS buffer -------
    float4 ra[4], rb[2];

    // prologue loads for chunk 0
#pragma unroll
    for (int p = 0; p < 4; ++p)
        ra[p] = *(const float4*)(Ab + (long long)(row0 + am0 + p * 32) * lda + akc);
    if (BT) {
#pragma unroll
        for (int p = 0; p < 2; ++p)
            rb[p] = *(const float4*)(Bb + (long long)(col0 + am0 + p * 32) * ldb + akc);
    } else {
#pragma unroll
        for (int p = 0; p < 2; ++p)
            rb[p] = *(const float4*)(Bb + (long long)(kk0 + p * 16) * ldb + col0 + nnc);
    }
    // commit chunk 0 to buffer 0
#pragma unroll
    for (int p = 0; p < 4; ++p) {
        float* sp = &As[0][(am0 + p * 32) * TS + akc];
        sp[0] = ra[p].x; sp[1] = ra[p].y; sp[2] = ra[p].z; sp[3] = ra[p].w;
    }
    if (BT) {
#pragma unroll
        for (int p = 0; p < 2; ++p) {
            float* sp = &Bs[0][(am0 + p * 32) * TS + akc];
            sp[0] = rb[p].x; sp[1] = rb[p].y; sp[2] = rb[p].z; sp[3] = rb[p].w;
        }
    } else {
#pragma unroll
        for (int p = 0; p < 2; ++p) {
            const int k = kk0 + p * 16;
            Bs[0][(nnc + 0) * TS + k] = rb[p].x;
            Bs[0][(nnc + 1) * TS + k] = rb[p].y;
            Bs[0][(nnc + 2) * TS + k] = rb[p].z;
            Bs[0][(nnc + 3) * TS + k] = rb[p].w;
        }
    }
    __syncthreads();

    for (int kt = 0; kt < K; kt += KC) {
        const bool more = (kt + KC) < K;
        if (more) {
            const int kn = kt + KC;
            // issue next chunk's global loads BEFORE compute (latency hiding)
#pragma unroll
            for (int p = 0; p < 4; ++p)
                ra[p] = *(const float4*)(Ab + (long long)(row0 + am0 + p * 32) * lda + kn + akc);
            if (BT) {
#pragma unroll
                for (int p = 0; p < 2; ++p)
                    rb[p] = *(const float4*)(Bb + (long long)(col0 + am0 + p * 32) * ldb + kn + akc);
            } else {
#pragma unroll
                for (int p = 0; p < 2; ++p)
                    rb[p] = *(const float4*)(Bb + (long long)(kn + kk0 + p * 16) * ldb + col0 + nnc);
            }
        }

        // ---- compute on current buffer ----
        {
            const float* __restrict__ Asb = As[cur];
            const float* __restrict__ Bsb = Bs[cur];
#pragma unroll
            for (int kk = 0; kk < KC; kk += 4) {
                const int ko = kk + 2 * half;
                const v2f a0 = *(const v2f*)&Asb[(wm * 32 +  0 + lm) * TS + ko];
                const v2f a1 = *(const v2f*)&Asb[(wm * 32 + 16 + lm) * TS + ko];
                const v2f b0 = *(const v2f*)&Bsb[(wn * 32 +  0 + lm) * TS + ko];
                const v2f b1 = *(const v2f*)&Bsb[(wn * 32 + 16 + lm) * TS + ko];
                acc[0][0] = __builtin_amdgcn_wmma_f32_16x16x4_f32(
                    false, a0, false, b0, (short)0, acc[0][0], false, false);
                acc[0][1] = __builtin_amdgcn_wmma_f32_16x16x4_f32(
                    false, a0, false, b1, (short)0, acc[0][1], false, false);
                acc[1][0] = __builtin_amdgcn_wmma_f32_16x16x4_f32(
                    false, a1, false, b0, (short)0, acc[1][0], false, false);
                acc[1][1] = __builtin_amdgcn_wmma_f32_16x16x4_f32(
                    false, a1, false, b1, (short)0, acc[1][1], false, false);
            }
        }

        if (more) {
            const int nxt = cur ^ 1;
#pragma unroll
            for (int p = 0; p < 4; ++p) {
                float* sp = &As[nxt][(am0 + p * 32) * TS + akc];
                sp[0] = ra[p].x; sp[1] = ra[p].y; sp[2] = ra[p].z; sp[3] = ra[p].w;
            }
            if (BT) {
#pragma unroll
                for (int p = 0; p < 2; ++p) {
                    float* sp = &Bs[nxt][(am0 + p * 32) * TS + akc];
                    sp[0] = rb[p].x; sp[1] = rb[p].y; sp[2] = rb[p].z; sp[3] = rb[p].w;
                }
            } else {
#pragma unroll
                for (int p = 0; p < 2; ++p) {
                    const int k = kk0 + p * 16;
                    Bs[nxt][(nnc + 0) * TS + k] = rb[p].x;
                    Bs[nxt][(nnc + 1) * TS + k] = rb[p].y;
                    Bs[nxt][(nnc + 2) * TS + k] = rb[p].z;
                    Bs[nxt][(nnc + 3) * TS + k] = rb[p].w;
                }
            }
            __syncthreads();
            cur = nxt;
        }
    }
#endif

    // --- epilogue: C VGPR r -> M=r (lanes 0-15) / M=r+8 (lanes 16-31) ---
#pragma unroll
    for (int i = 0; i < 2; ++i) {
#pragma unroll
        for (int j = 0; j < 2; ++j) {
            const int colg = col0 + wn * 32 + j * 16 + lm;
            const float bv = bias ? bias[colg] : 0.0f;
#pragma unroll
            for (int r = 0; r < 8; ++r) {
                const int rowg = row0 + wm * 32 + i * 16 + r + 8 * half;
                Cb[(long long)rowg * ldc + colg] = acc[i][j][r] * alpha + bv;
            }
        }
    }
}

// ---------------- row-wise helper kernels (1024-wide rows) -----------------

__device__ __forceinline__ float block_reduce_sum256(float v) {
    __shared__ float sd[256];
    const int tid = threadIdx.x;
    __syncthreads();
    sd[tid] = v;
    __syncthreads();
#pragma unroll
    for (int s = 128; s > 0; s >>= 1) {
        if (tid < s) sd[tid] += sd[tid + s];
        __syncthreads();
    }
    return sd[0];
}

__device__ __forceinline__ float block_reduce_max256(float v) {
    __shared__ float sd[256];
    const int tid = threadIdx.x;
    __syncthreads();
    sd[tid] = v;
    __syncthreads();
#pragma unroll
    for (int s = 128; s > 0; s >>= 1) {
        if (tid < s) sd[tid] = fmaxf(sd[tid], sd[tid + s]);
        __syncthreads();
    }
    return sd[0];
}

// enhanced = 0.9*enhanced + 0.1*row_mean(enhanced)
__global__ __launch_bounds__(256) void mean_blend_rows(float* __restrict__ E) {
    float* p = E + (long long)blockIdx.x * 1024;
    const int tid = threadIdx.x;
    float x[4], s = 0.0f;
#pragma unroll
    for (int i = 0; i < 4; ++i) { x[i] = p[tid + 256 * i]; s += x[i]; }
    const float mean = block_reduce_sum256(s) * (1.0f / 1024.0f);
#pragma unroll
    for (int i = 0; i < 4; ++i) p[tid + 256 * i] = 0.9f * x[i] + 0.1f * mean;
}

// enhanced += 0.05 * corr / (||corr_row||_2 + 1e-8)
__global__ __launch_bounds__(256) void norm_add_rows(float* __restrict__ E,
                                                     const float* __restrict__ Cr) {
    const long long row = blockIdx.x;
    const float* c = Cr + row * 1024;
    float*       e = E  + row * 1024;
    const int tid = threadIdx.x;
    float cv[4], ss = 0.0f;
#pragma unroll
    for (int i = 0; i < 4; ++i) { cv[i] = c[tid + 256 * i]; ss += cv[i] * cv[i]; }
    const float tot = block_reduce_sum256(ss);
    const float inv = 0.05f / (sqrtf(tot) + 1e-8f);
#pragma unroll
    for (int i = 0; i < 4; ++i) e[tid + 256 * i] += cv[i] * inv;
}

__global__ __launch_bounds__(256) void softmax_rows(float* __restrict__ E) {
    float* p = E + (long long)blockIdx.x * 1024;
    const int tid = threadIdx.x;
    float x[4], mx = -3.402823466e38f;
#pragma unroll
    for (int i = 0; i < 4; ++i) { x[i] = p[tid + 256 * i]; mx = fmaxf(mx, x[i]); }
    mx = block_reduce_max256(mx);
    float s = 0.0f;
#pragma unroll
    for (int i = 0; i < 4; ++i) { x[i] = __expf(x[i] - mx); s += x[i]; }
    s = block_reduce_sum256(s);
    const float inv = 1.0f / s;
#pragma unroll
    for (int i = 0; i < 4; ++i) p[tid + 256 * i] = x[i] * inv;
}

// ---------------------------------------------------------------------------

extern "C" void kernel_launch(void* const* d_in, const int* in_sizes, int n_in,
                              void* d_out, int out_size, void* d_ws, size_t ws_size,
                              hipStream_t stream) {
    (void)in_sizes; (void)n_in; (void)out_size; (void)ws_size;
    const float* x      = (const float*)d_in[0];  // [4,1024,512]
    const float* w_qkv  = (const float*)d_in[1];  // [1536,512]
    const float* w_proj = (const float*)d_in[2];  // [512,512]
    const float* b_proj = (const float*)d_in[3];  // [512]
    float* out = (float*)d_out;                    // [4,1024,512]

    const long long B = 4, N = 1024, Cc = 512, H = 8;
    const long long BN = B * N;                    // 4096
    const long long NN_ = N * N;                   // 1048576

    // workspace carve-out (floats): qkv 24MB, E 128MB, corr 128MB, obuf 8MB
    float* qkvb = (float*)d_ws;                    // [4096][1536] as [B,N,3,H,Dh]
    float* E    = qkvb + BN * 3 * Cc;              // [32][1024][1024]
    float* Cr   = E + B * H * NN_;                 // [32][1024][1024]
    float* obuf = Cr + B * H * NN_;                // [4096][512] as [B,N,H,Dh]

    const dim3 blk(256);

    // 1) qkv = x @ w_qkv^T   (NT, 4096x1536x512)
    gemm_wmma_f32<true><<<dim3(1536 / TILE_N, 4096 / TILE_M, 1), blk, 0, stream>>>(
        x, 512, 0, 0, w_qkv, 512, 0, 0, qkvb, 1536, 0, 0,
        4096, 1536, 512, 1, 1.0f, nullptr);

    // 2) E = (q @ k^T) * Dh^-0.5   per head (NT, 32 x 1024x1024x64)
    gemm_wmma_f32<true><<<dim3(1024 / TILE_N, 1024 / TILE_M, 32), blk, 0, stream>>>(
        qkvb,       1536, N * 3 * Cc, 64,      // q slice: [b][n][0][h][:]
        qkvb + 512, 1536, N * 3 * Cc, 64,      // k slice: [b][n][1][h][:]
        E, 1024, H * NN_, NN_,
        1024, 1024, 64, (int)H, 0.125f, nullptr);

    // 3) enhancement loop (depth 3)
    for (int it = 0; it < 3; ++it) {
        mean_blend_rows<<<32768, blk, 0, stream>>>(E);
        // corr = E @ E^T per head (NT, 32 x 1024^3) -- dominant FLOPs
        gemm_wmma_f32<true><<<dim3(1024 / TILE_N, 1024 / TILE_M, 32), blk, 0, stream>>>(
            E, 1024, H * NN_, NN_,
            E, 1024, H * NN_, NN_,
            Cr, 1024, H * NN_, NN_,
            1024, 1024, 1024, (int)H, 1.0f, nullptr);
        norm_add_rows<<<32768, blk, 0, stream>>>(E, Cr);
    }

    // 4) softmax over rows
    softmax_rows<<<32768, blk, 0, stream>>>(E);

    // 5) obuf = attn @ v   per head (NN, 32 x 1024x64x1024)
    gemm_wmma_f32<false><<<dim3(64 / TILE_N, 1024 / TILE_M, 32), blk, 0, stream>>>(
        E, 1024, H * NN_, NN_,
        qkvb + 1024, 1536, N * 3 * Cc, 64,     // v slice: [b][m][2][h][:]
        obuf, 512, N * Cc, 64,                 // out slice: [b][n][h][:]
        1024, 64, 1024, (int)H, 1.0f, nullptr);

    // 6) out = obuf @ w_proj^T + b_proj  (NT, 4096x512x512)
    gemm_wmma_f32<true><<<dim3(512 / TILE_N, 4096 / TILE_M, 1), blk, 0, stream>>>(
        obuf, 512, 0, 0, w_proj, 512, 0, 0, out, 512, 0, 0,
        4096, 512, 512, 1, 1.0f, b_proj);
}